// LocalExtractor_31688268710053
// MI455X (gfx1250) — compile-verified
//
#include <hip/hip_runtime.h>

typedef float v2f __attribute__((ext_vector_type(2)));
typedef float v8f __attribute__((ext_vector_type(8)));

#define NP 102400   // points = 320*320
#define DIM 64      // feature dim / number of output channels
#define KIN 256     // input channels
#define NC 64       // clusters

// ---------------------------------------------------------------------------
// Kernel 1: pts[p, o] = sum_i conv_w[o,i]*x[i,p] + conv_b[o]
// One wave owns a 16-point x 64-channel tile: A (16x4 f32) loaded once per
// K-step, reused for 4 B tiles -> x is read exactly once from HBM.
// ---------------------------------------------------------------------------
__global__ __launch_bounds__(256) void conv_wmma_kernel(
    const float* __restrict__ x, const float* __restrict__ w,
    const float* __restrict__ bias, float* __restrict__ pts)
{
  const int wave = blockIdx.x * 8 + (threadIdx.x >> 5);   // 6400 waves total
  const int lane = threadIdx.x & 31;
  const int grp  = lane >> 4;       // half-wave: 0 -> K 0/1 (rows 0-7), 1 -> K 2/3 (rows 8-15)
  const int l16  = lane & 15;
  const int p0   = wave * 16;

  v8f acc[4] = {};
  for (int k0 = 0; k0 < KIN; k0 += 4) {
    const int ka = k0 + grp * 2;
    v2f a;
    a.x = x[(size_t)ka       * NP + p0 + l16];   // A[m=l16, k=ka]
    a.y = x[(size_t)(ka + 1) * NP + p0 + l16];   // A[m=l16, k=ka+1]
#pragma unroll
    for (int nt = 0; nt < 4; ++nt) {
      const int n = nt * 16 + l16;
      v2f b = *(const v2f*)(w + n * KIN + ka);   // B[k,n] = w[n, k] (contiguous pair)
      acc[nt] = __builtin_amdgcn_wmma_f32_16x16x4_f32(
          false, a, false, b, (short)0, acc[nt], false, false);
    }
  }
#pragma unroll
  for (int nt = 0; nt < 4; ++nt) {
    const int n = nt * 16 + l16;
    const float bb = bias[n];
#pragma unroll
    for (int v = 0; v < 8; ++v) {
      const int m = v + grp * 8;                 // C/D row layout
      pts[(size_t)(p0 + m) * DIM + n] = acc[nt][v] + bb;
    }
  }
}

// ---------------------------------------------------------------------------
// Kernel 2: centroid init: cent[c] = pts[c], csq2[c] = 0.5*||cent[c]||^2
// ---------------------------------------------------------------------------
__global__ __launch_bounds__(64) void init_cent_kernel(
    const float* __restrict__ pts, float* __restrict__ cent, float* __restrict__ csq2)
{
  const int c = blockIdx.x, t = threadIdx.x;
  const float v = pts[c * DIM + t];
  cent[c * DIM + t] = v;
  __shared__ float red[64];
  red[t] = v * v; __syncthreads();
  for (int s = 32; s > 0; s >>= 1) { if (t < s) red[t] += red[t + s]; __syncthreads(); }
  if (t == 0) csq2[c] = 0.5f * red[0];
}

// ---------------------------------------------------------------------------
// Kernel 3: assign. score[p,c] = dot(pts[p],cent[c]) - 0.5*||cent[c]||^2.
// argmin(dist) == argmax(score), ties -> lowest c. WMMA with C preloaded.
// ---------------------------------------------------------------------------
__global__ __launch_bounds__(256) void assign_wmma_kernel(
    const float* __restrict__ pts, const float* __restrict__ cent,
    const float* __restrict__ csq2, int* __restrict__ labels)
{
  const int wave = blockIdx.x * 8 + (threadIdx.x >> 5);
  const int lane = threadIdx.x & 31;
  const int grp  = lane >> 4;
  const int l16  = lane & 15;
  const int p0   = wave * 16;

  v8f acc[4];
#pragma unroll
  for (int nt = 0; nt < 4; ++nt) {
    const float c0 = -csq2[nt * 16 + l16];
#pragma unroll
    for (int v = 0; v < 8; ++v) acc[nt][v] = c0;
  }

  for (int k0 = 0; k0 < DIM; k0 += 4) {
    const int ka = k0 + grp * 2;
    v2f a = *(const v2f*)(pts + (size_t)(p0 + l16) * DIM + ka);
#pragma unroll
    for (int nt = 0; nt < 4; ++nt) {
      v2f b = *(const v2f*)(cent + (nt * 16 + l16) * DIM + ka);
      acc[nt] = __builtin_amdgcn_wmma_f32_16x16x4_f32(
          false, a, false, b, (short)0, acc[nt], false, false);
    }
  }

  // per-row argmax across 16 lanes (each half-wave owns rows grp*8 .. grp*8+7)
  float bestv[8]; int besti[8];
#pragma unroll
  for (int v = 0; v < 8; ++v) { bestv[v] = -3.4e38f; besti[v] = 0; }
#pragma unroll
  for (int nt = 0; nt < 4; ++nt) {
#pragma unroll
    for (int v = 0; v < 8; ++v) {
      float val = acc[nt][v];
      int   idx = nt * 16 + l16;
#pragma unroll
      for (int off = 8; off > 0; off >>= 1) {
        const float ov = __shfl_xor(val, off, 16);
        const int   oi = __shfl_xor(idx, off, 16);
        if (ov > val || (ov == val && oi < idx)) { val = ov; idx = oi; }
      }
      if (val > bestv[v] || (val == bestv[v] && idx < besti[v])) { bestv[v] = val; besti[v] = idx; }
    }
  }
  if (l16 == 0) {
#pragma unroll
    for (int v = 0; v < 8; ++v) labels[p0 + grp * 8 + v] = besti[v];
  }
}

// ---------------------------------------------------------------------------
// Kernel 4: zero accumulators
// ---------------------------------------------------------------------------
__global__ void zero_kernel(float* __restrict__ sums, float* __restrict__ counts)
{
  const int i = blockIdx.x * blockDim.x + threadIdx.x;
  if (i < NC * DIM) sums[i] = 0.f;
  if (i < NC)       counts[i] = 0.f;
}

// ---------------------------------------------------------------------------
// Kernel 5: segment-sum via LDS atomics, one global flush per block.
// ---------------------------------------------------------------------------
__global__ __launch_bounds__(256) void accum_kernel(
    const float* __restrict__ pts, const int* __restrict__ labels,
    float* __restrict__ sums, float* __restrict__ counts)
{
  __shared__ float lsum[NC * DIM];
  __shared__ float lcnt[NC];
  const int tid = threadIdx.x;
  for (int i = tid; i < NC * DIM; i += 256) lsum[i] = 0.f;
  if (tid < NC) lcnt[tid] = 0.f;
  __syncthreads();

  const int p = blockIdx.x * 256 + tid;
  const int l = labels[p];
  atomicAdd(&lcnt[l], 1.0f);
  const float* row = pts + (size_t)p * DIM;
  for (int j = 0; j < DIM; ++j) atomicAdd(&lsum[l * DIM + j], row[j]);
  __syncthreads();

  for (int i = tid; i < NC * DIM; i += 256) atomicAdd(&sums[i], lsum[i]);
  if (tid < NC) atomicAdd(&counts[tid], lcnt[tid]);
}

// ---------------------------------------------------------------------------
// Kernel 6: centroid update + csq2 refresh
// ---------------------------------------------------------------------------
__global__ __launch_bounds__(64) void finalize_cent_kernel(
    const float* __restrict__ sums, const float* __restrict__ counts,
    float* __restrict__ cent, float* __restrict__ csq2)
{
  const int c = blockIdx.x, t = threadIdx.x;
  const float cnt = counts[c];
  const float oldv = cent[c * DIM + t];
  const float nv = (cnt > 0.0f) ? (sums[c * DIM + t] / fmaxf(cnt, 1.0f)) : oldv;
  cent[c * DIM + t] = nv;
  __shared__ float red[64];
  red[t] = nv * nv; __syncthreads();
  for (int s = 32; s > 0; s >>= 1) { if (t < s) red[t] += red[t + s]; __syncthreads(); }
  if (t == 0) csq2[c] = 0.5f * red[0];
}

// ---------------------------------------------------------------------------
// Kernel 7: one-hot masks + 2x2 mean pool fused.
// ---------------------------------------------------------------------------
__global__ __launch_bounds__(256) void pool_kernel(
    const int* __restrict__ labels, float* __restrict__ out)
{
  const int idx = blockIdx.x * 256 + threadIdx.x;   // 160*160 = 25600
  const int wo = idx % 160, ho = idx / 160;
  const int h = ho * 2, w = wo * 2;
  const int l00 = labels[h * 320 + w];
  const int l01 = labels[h * 320 + w + 1];
  const int l10 = labels[(h + 1) * 320 + w];
  const int l11 = labels[(h + 1) * 320 + w + 1];
#pragma unroll 8
  for (int c = 0; c < NC; ++c) {
    const float v = 0.25f * (float)((l00 == c) + (l01 == c) + (l10 == c) + (l11 == c));
    out[c * 25600 + idx] = v;
  }
}

// ---------------------------------------------------------------------------
extern "C" void kernel_launch(void* const* d_in, const int* in_sizes, int n_in,
                              void* d_out, int out_size, void* d_ws, size_t ws_size,
                              hipStream_t stream)
{
  (void)in_sizes; (void)n_in; (void)out_size; (void)ws_size;
  const float* x = (const float*)d_in[0];   // (256, 320, 320)
  const float* w = (const float*)d_in[1];   // (64, 256)
  const float* b = (const float*)d_in[2];   // (64,)
  float* out = (float*)d_out;               // (64, 160, 160)

  float* pts    = (float*)d_ws;             // NP*DIM
  float* cent   = pts + (size_t)NP * DIM;   // NC*DIM
  float* csq2   = cent + NC * DIM;          // NC
  float* sums   = csq2 + NC;                // NC*DIM
  float* counts = sums + NC * DIM;          // NC
  int*   labels = (int*)(counts + NC);      // NP

  conv_wmma_kernel<<<800, 256, 0, stream>>>(x, w, b, pts);
  init_cent_kernel<<<NC, 64, 0, stream>>>(pts, cent, csq2);
  for (int it = 0; it < 10; ++it) {
    assign_wmma_kernel<<<800, 256, 0, stream>>>(pts, cent, csq2, labels);
    zero_kernel<<<16, 256, 0, stream>>>(sums, counts);
    accum_kernel<<<400, 256, 0, stream>>>(pts, labels, sums, counts);
    finalize_cent_kernel<<<NC, 64, 0, stream>>>(sums, counts, cent, csq2);
  }
  assign_wmma_kernel<<<800, 256, 0, stream>>>(pts, cent, csq2, labels);
  pool_kernel<<<100, 256, 0, stream>>>(labels, out);
}